// RNN_64501818851829
// MI455X (gfx1250) — compile-verified
//
#include <hip/hip_runtime.h>

typedef __attribute__((ext_vector_type(2))) float v2f;
typedef __attribute__((ext_vector_type(8))) float v8f;

#define B_TOT 4096
#define T_TOT 1024
#define KC 44   // padded K for combined (41 -> 44), row stride in floats
#define KP1 11  // K-chunks (of 4) for layer 1
#define KA 42   // padded row stride for a/o scratch (K=40, even for b64 align)
#define KP2 10  // K-chunks for layer 2
#define NP 48   // padded N (40 -> 48) for weight fragment arrays

// B-matrix fragment (K=4 x N=16) from LDS pair layout:
// element (p, n) holds {B[2p][n], B[2p+1][n]} as a float2.
__device__ __forceinline__ v2f ld_b_frag(const float* base, int npad, int c, int n0, int lane) {
  int p = 2 * c + (lane >> 4);
  int n = n0 + (lane & 15);
  return *(const v2f*)(base + (p * npad + n) * 2);
}

// A-matrix fragment (M=16 x K=4) from a row-major [16][stride] LDS tile:
// lane m (lanes 0-15) holds K = {4c, 4c+1}; lanes 16-31 hold K = {4c+2, 4c+3}.
__device__ __forceinline__ v2f ld_a_frag(const float* rows, int stride, int c, int lane) {
  int m = lane & 15;
  return *(const v2f*)(rows + m * stride + 4 * c + 2 * (lane >> 4));
}

__global__ __launch_bounds__(128, 1) void rnn_wmma_f32(
    const float* __restrict__ inp,
    const float* __restrict__ w1h, const float* __restrict__ b1h,
    const float* __restrict__ w2h, const float* __restrict__ b2h,
    const float* __restrict__ w1o, const float* __restrict__ b1o,
    const float* __restrict__ w2o, const float* __restrict__ b2o,
    float* __restrict__ outp)
{
  // Weight fragments (shared by all 4 waves, staged once).
  __shared__ __align__(16) float sW1h[22 * NP * 2];
  __shared__ __align__(16) float sW1o[22 * NP * 2];
  __shared__ __align__(16) float sW2h[20 * NP * 2];
  __shared__ __align__(16) float sW2o[20 * 16 * 2];
  // Per-wave recurrence scratch.
  __shared__ __align__(16) float sCB[4][16 * KC];
  __shared__ __align__(16) float sA [4][16 * KA];
  __shared__ __align__(16) float sO [4][16 * KA];

  const int tid  = threadIdx.x;
  const int lane = tid & 31;
  const int wave = tid >> 5;
  const int nl   = lane & 15;
  const int half = lane >> 4;

  // --- Stage layer-1 weights: B[k][n] = W1[n][k], K padded 41->44, N padded 40->48.
  for (int idx = tid; idx < 22 * NP; idx += 128) {
    int p = idx / NP, n = idx % NP;
    int k0 = 2 * p, k1 = 2 * p + 1;
    float h0 = 0.f, h1 = 0.f, o0 = 0.f, o1 = 0.f;
    if (n < 40) {
      if (k0 < 41) { h0 = w1h[n * 41 + k0]; o0 = w1o[n * 41 + k0]; }
      if (k1 < 41) { h1 = w1h[n * 41 + k1]; o1 = w1o[n * 41 + k1]; }
    }
    sW1h[idx * 2] = h0; sW1h[idx * 2 + 1] = h1;
    sW1o[idx * 2] = o0; sW1o[idx * 2 + 1] = o1;
  }
  // --- Layer-2 hidden weights: B[k][h] = W2h[h][k], K=40, N padded to 48.
  for (int idx = tid; idx < 20 * NP; idx += 128) {
    int p = idx / NP, n = idx % NP;
    float h0 = 0.f, h1 = 0.f;
    if (n < 40) { h0 = w2h[n * 40 + 2 * p]; h1 = w2h[n * 40 + 2 * p + 1]; }
    sW2h[idx * 2] = h0; sW2h[idx * 2 + 1] = h1;
  }
  // --- Output weights: B[k][0] = W2o[k], N padded to 16 (cols 1..15 zero).
  for (int idx = tid; idx < 20 * 16; idx += 128) {
    int p = idx / 16, n = idx % 16;
    sW2o[idx * 2]     = (n == 0) ? w2o[2 * p]     : 0.f;
    sW2o[idx * 2 + 1] = (n == 0) ? w2o[2 * p + 1] : 0.f;
  }

  float* cb = sCB[wave];
  float* aw = sA[wave];
  float* ow = sO[wave];
  // hidden0 = 0 and K-padding columns (41..43) = 0 forever.
  for (int i = lane; i < 16 * KC; i += 32) cb[i] = 0.f;
  __syncthreads();  // only barrier in the kernel

  const int b0 = (blockIdx.x * 4 + wave) * 16;

  // Per-lane bias values in D-layout (lane's N column, same for all 8 rows).
  float rb1h[3], rb1o[3], rb2h[3];
  #pragma unroll
  for (int nt = 0; nt < 3; ++nt) {
    int n = nt * 16 + nl;
    rb1h[nt] = (n < 40) ? b1h[n] : 0.f;
    rb1o[nt] = (n < 40) ? b1o[n] : 0.f;
    rb2h[nt] = (n < 40) ? b2h[n] : 0.f;
  }
  const float rb2o = b2o[0];

  // Prime input: lane m (< 16) carries u_t for batch row b0+m.
  float u = 0.f;
  if (lane < 16) u = inp[(size_t)(b0 + lane) * T_TOT];

  for (int t = 0; t < T_TOT; ++t) {
    // A fragments of combined = [u | hidden]; u injected in-register (col 0).
    v2f ca[KP1];
    #pragma unroll
    for (int c = 0; c < KP1; ++c) ca[c] = ld_a_frag(cb, KC, c, lane);
    if (lane < 16) ca[0].x = u;

    // Prefetch next timestep's input (overlaps with compute below).
    float unext = 0.f;
    if (lane < 16 && (t + 1) < T_TOT)
      unext = inp[(size_t)(b0 + lane) * T_TOT + (t + 1)];

    // ---- Layer 1: a = leaky(combined @ W1h^T + b1h), o = leaky(... W1o ...)
    #pragma unroll
    for (int nt = 0; nt < 3; ++nt) {
      const int n = nt * 16 + nl;
      float bh = rb1h[nt];
      v8f acc = {bh, bh, bh, bh, bh, bh, bh, bh};
      #pragma unroll
      for (int c = 0; c < KP1; ++c)
        acc = __builtin_amdgcn_wmma_f32_16x16x4_f32(
            false, ca[c], false, ld_b_frag(sW1h, NP, c, nt * 16, lane),
            (short)0, acc, false, false);
      #pragma unroll
      for (int i = 0; i < 8; ++i) {
        float x = acc[i];
        x = fmaxf(x, 0.01f * x);                 // LeakyReLU, slope < 1
        if (n < 40) aw[(half * 8 + i) * KA + n] = x;
      }
      float bo = rb1o[nt];
      v8f acco = {bo, bo, bo, bo, bo, bo, bo, bo};
      #pragma unroll
      for (int c = 0; c < KP1; ++c)
        acco = __builtin_amdgcn_wmma_f32_16x16x4_f32(
            false, ca[c], false, ld_b_frag(sW1o, NP, c, nt * 16, lane),
            (short)0, acco, false, false);
      #pragma unroll
      for (int i = 0; i < 8; ++i) {
        float x = acco[i];
        x = fmaxf(x, 0.01f * x);
        if (n < 40) ow[(half * 8 + i) * KA + n] = x;
      }
    }

    // ---- Layer 2 A fragments (LDS transpose of the D-layout results).
    v2f aa[KP2], oa[KP2];
    #pragma unroll
    for (int c = 0; c < KP2; ++c) {
      aa[c] = ld_a_frag(aw, KA, c, lane);
      oa[c] = ld_a_frag(ow, KA, c, lane);
    }

    // hidden_new = a @ W2h^T + b2h  -> combined columns 1..40
    #pragma unroll
    for (int nt = 0; nt < 3; ++nt) {
      const int n = nt * 16 + nl;
      float bh = rb2h[nt];
      v8f acc = {bh, bh, bh, bh, bh, bh, bh, bh};
      #pragma unroll
      for (int c = 0; c < KP2; ++c)
        acc = __builtin_amdgcn_wmma_f32_16x16x4_f32(
            false, aa[c], false, ld_b_frag(sW2h, NP, c, nt * 16, lane),
            (short)0, acc, false, false);
      #pragma unroll
      for (int i = 0; i < 8; ++i)
        if (n < 40) cb[(half * 8 + i) * KC + 1 + n] = acc[i];
    }

    // out[:, t] = o @ W2o^T + b2o  (only N=0 column is real)
    {
      v8f acc = {rb2o, rb2o, rb2o, rb2o, rb2o, rb2o, rb2o, rb2o};
      #pragma unroll
      for (int c = 0; c < KP2; ++c)
        acc = __builtin_amdgcn_wmma_f32_16x16x4_f32(
            false, oa[c], false, ld_b_frag(sW2o, 16, c, 0, lane),
            (short)0, acc, false, false);
      if (nl == 0) {   // lanes 0 and 16 hold rows 0..7 / 8..15 of column 0
        #pragma unroll
        for (int i = 0; i < 8; ++i)
          outp[(size_t)(b0 + half * 8 + i) * T_TOT + t] = acc[i];
      }
    }

    u = unext;
  }
}

extern "C" void kernel_launch(void* const* d_in, const int* in_sizes, int n_in,
                              void* d_out, int out_size, void* d_ws, size_t ws_size,
                              hipStream_t stream) {
  const float* inp = (const float*)d_in[0];
  const float* w1h = (const float*)d_in[1];
  const float* b1h = (const float*)d_in[2];
  const float* w2h = (const float*)d_in[3];
  const float* b2h = (const float*)d_in[4];
  const float* w1o = (const float*)d_in[5];
  const float* b1o = (const float*)d_in[6];
  const float* w2o = (const float*)d_in[7];
  const float* b2o = (const float*)d_in[8];
  float* outp = (float*)d_out;
  (void)in_sizes; (void)n_in; (void)out_size; (void)d_ws; (void)ws_size;

  // 256 batch tiles of 16 rows; 4 independent waves (tiles) per 128-thread block.
  rnn_wmma_f32<<<B_TOT / 64, 128, 0, stream>>>(
      inp, w1h, b1h, w2h, b2h, w1o, b1o, w2o, b2o, outp);
}